// HierarchyModel_21199958573825
// MI455X (gfx1250) — compile-verified
//
#include <hip/hip_runtime.h>
#include <math.h>

typedef __attribute__((ext_vector_type(2))) float v2f;
typedef __attribute__((ext_vector_type(8))) float v8f;

#define NN   1024
#define DD   16
#define CC   8192
#define OD   128
#define EPSF 1e-15f
#define PIF  3.14159265358979323846f

// ---------------- init: zero accumulators + colSum ----------------
__global__ void init_kernel(float* ws) {
    int idx = blockIdx.x * blockDim.x + threadIdx.x;
    if (idx < 8 + NN) ws[idx] = 0.0f;   // acc[0..7], colSum[0..1023]
}

// ---------------- per-node small losses (exceed, shape, positive) ----------------
__global__ __launch_bounds__(256) void node_loss_kernel(
    const int* __restrict__ idIdx, const int* __restrict__ parIdx,
    const float* __restrict__ childEmb, const float* __restrict__ res,
    const float* __restrict__ leaves, float* __restrict__ acc) {
    __shared__ float sred[256];
    int t = threadIdx.x;
    int idx = blockIdx.x * blockDim.x + t;           // over N*D = 16384
    float e = 0.f, sh = 0.f, po = 0.f;
    if (idx < NN * DD) {
        int n = idx >> 4, d = idx & 15;
        int id = idIdx[n], par = parIdx[n];
        float cl = childEmb[id * 32 + d];
        float ch = childEmb[id * 32 + 16 + d];
        float rl = res[par * 32 + d];
        float rh = res[par * 32 + 16 + d];
        float pl = rl + 1.0f, ph = rh + 1.0f;
        e = fmaxf(pl - cl, 0.f) + fmaxf(ch - ph, 0.f);
        float diff = ch - cl;
        float pdiff = rh - rl;
        float numer = fmaxf(diff / pdiff, EPSF);
        float sdiv = fminf(numer / leaves[id], PIF * 0.49999f);
        sh = fminf(fabsf(tanf((sdiv - 1.0f) * (PIF * 0.5f))), 10000.0f);
        po = fmaxf(expf(-diff), EPSF);
    }
    sred[t] = e; __syncthreads();
    for (int s = 128; s > 0; s >>= 1) { if (t < s) sred[t] += sred[t + s]; __syncthreads(); }
    if (t == 0) atomicAdd(&acc[0], sred[0]);
    __syncthreads();
    sred[t] = sh; __syncthreads();
    for (int s = 128; s > 0; s >>= 1) { if (t < s) sred[t] += sred[t + s]; __syncthreads(); }
    if (t == 0) atomicAdd(&acc[1], sred[0]);
    __syncthreads();
    sred[t] = po; __syncthreads();
    for (int s = 128; s > 0; s >>= 1) { if (t < s) sred[t] += sred[t + s]; __syncthreads(); }
    if (t == 0) atomicAdd(&acc[2], sred[0]);
}

// ---------------- inner[n] = sum_k omega[n,k]^2 ----------------
__global__ void inner_kernel(const float* __restrict__ omega, float* __restrict__ inner) {
    int n = blockIdx.x * blockDim.x + threadIdx.x;
    if (n < NN) {
        float s = 0.f;
        #pragma unroll 8
        for (int k = 0; k < OD; ++k) { float v = omega[n * OD + k]; s += v * v; }
        inner[n] = s;
    }
}

// ---------------- omegaDist via V_WMMA_F32_16X16X4_F32 ----------------
// G = omega @ omega^T (1024x1024x128 f32); omegaDist = max(inner_i+inner_j-2G, EPS)
__global__ __launch_bounds__(32) void omega_dist_kernel(
    const float* __restrict__ omega, const float* __restrict__ inner,
    float* __restrict__ omegaMat, float* __restrict__ acc) {
    const int j0 = blockIdx.x * 16;
    const int i0 = blockIdx.y * 16;
    const int lane = threadIdx.x;          // 0..31, one wave, EXEC all ones
    const int m    = lane & 15;
    const int koff = (lane >> 4) * 2;      // lanes 0-15 -> K{0,1}; 16-31 -> K{2,3}
    const float* aRow = omega + (size_t)(i0 + m) * OD + koff;
    const float* bRow = omega + (size_t)(j0 + m) * OD + koff;
    v8f c = {};
    #pragma unroll 4
    for (int kc = 0; kc < OD; kc += 4) {
        v2f a; a.x = aRow[kc]; a.y = aRow[kc + 1];
        v2f b; b.x = bRow[kc]; b.y = bRow[kc + 1];
        c = __builtin_amdgcn_wmma_f32_16x16x4_f32(
                false, a, false, b, (short)0, c, false, false);
    }
    // D layout: c[r] -> row (r + 8*(lane>>4)), col (lane&15) of the tile
    const int col = j0 + (lane & 15);
    const float innJ = inner[col];
    float ss = 0.f;
    #pragma unroll
    for (int r = 0; r < 8; ++r) {
        int row = i0 + r + 8 * (lane >> 4);
        float v = fmaxf(inner[row] + innJ - 2.0f * c[r], EPSF);
        omegaMat[(size_t)row * NN + col] = v;
        ss += v * v;
    }
    for (int off = 16; off > 0; off >>= 1) ss += __shfl_down(ss, off, 32);
    if (lane == 0) atomicAdd(&acc[5], ss);
}

// ---------------- N x N pair pass: realDist, colSum, ||real||^2, lossOverlap ----------------
__global__ __launch_bounds__(256) void pair_kernel(
    const int* __restrict__ idIdx,
    const float* __restrict__ childEmb,
    const float* __restrict__ layerDist,
    float* __restrict__ colSum, float* __restrict__ acc) {
    __shared__ float sCLi[16][16], sCHi[16][16], sDInv[16][16];
    __shared__ float sCLj[16][16], sCHj[16][16];
    __shared__ int   sIdI[16], sIdJ[16];
    __shared__ float sRed[256];
    const int i0 = blockIdx.y * 16, j0 = blockIdx.x * 16;
    const int t = threadIdx.x;
    const int r = t >> 4, d = t & 15;     // also (iL, jL) in compute phase
    if (t < 16) sIdI[t] = idIdx[i0 + t];
    else if (t < 32) sIdJ[t - 16] = idIdx[j0 + t - 16];
    __syncthreads();
    {
        int gi = sIdI[r];
        float lo = childEmb[gi * 32 + d];
        float hi = childEmb[gi * 32 + 16 + d];
        sCLi[r][d] = lo; sCHi[r][d] = hi;
        sDInv[r][d] = 1.0f / fmaxf(hi - lo, EPSF);
        int gj = sIdJ[r];
        sCLj[r][d] = childEmb[gj * 32 + d];
        sCHj[r][d] = childEmb[gj * 32 + 16 + d];
    }
    __syncthreads();
    const int gI = i0 + r, gJ = j0 + d;
    float cur = 0.f, ov = 0.f;
    #pragma unroll
    for (int k = 0; k < DD; ++k) {
        float li = sCLi[r][k], lj = sCLj[d][k];
        cur += fabsf(li - lj);
        float o = fminf(sCHi[r][k], sCHj[d][k]) - fmaxf(li, lj);
        ov += fmaxf(o, 0.f) * sDInv[r][k];
    }
    if (gI == gJ) ov = 0.f;                                  // offdiag mask
    float real = layerDist[(size_t)sIdI[r] * CC + sIdJ[d]] + cur;
    // column sums (reduce over iL = r)
    sRed[t] = real; __syncthreads();
    for (int s = 8; s > 0; s >>= 1) {
        if (r < s) sRed[r * 16 + d] += sRed[(r + s) * 16 + d];
        __syncthreads();
    }
    if (r == 0) atomicAdd(&colSum[j0 + d], sRed[d]);
    __syncthreads();
    sRed[t] = real * real; __syncthreads();
    for (int s = 128; s > 0; s >>= 1) { if (t < s) sRed[t] += sRed[t + s]; __syncthreads(); }
    if (t == 0) atomicAdd(&acc[4], sRed[0]);
    __syncthreads();
    sRed[t] = ov; __syncthreads();
    for (int s = 128; s > 0; s >>= 1) { if (t < s) sRed[t] += sRed[t + s]; __syncthreads(); }
    if (t == 0) atomicAdd(&acc[3], sRed[0]);
}

// ---------------- pass 2: ||omegaNormed - distNormed||^2 ----------------
__global__ __launch_bounds__(256) void dist_norm_kernel(
    const float* __restrict__ omegaMat, const float* __restrict__ colSum,
    float* acc) {
    __shared__ float sred[256];
    const float invR = 1.0f / fmaxf(sqrtf(acc[4]), EPSF);
    const float invO = 1.0f / fmaxf(sqrtf(acc[5]), EPSF);
    const int t = threadIdx.x;
    float ss = 0.f;
    for (int idx = blockIdx.x * blockDim.x + t; idx < NN * NN;
         idx += gridDim.x * blockDim.x) {
        int j = idx & (NN - 1);
        float v = omegaMat[idx] * invO - colSum[j] * invR;
        ss += v * v;
    }
    sred[t] = ss; __syncthreads();
    for (int s = 128; s > 0; s >>= 1) { if (t < s) sred[t] += sred[t + s]; __syncthreads(); }
    if (t == 0) atomicAdd(&acc[6], sred[0]);
}

// ---------------- combine ----------------
__global__ void final_kernel(const float* __restrict__ acc, float* __restrict__ out) {
    if (threadIdx.x == 0)
        out[0] = sqrtf(acc[6]) + acc[0] + acc[1] + acc[2] + acc[3];
}

extern "C" void kernel_launch(void* const* d_in, const int* in_sizes, int n_in,
                              void* d_out, int out_size, void* d_ws, size_t ws_size,
                              hipStream_t stream) {
    const int*   idIdx    = (const int*)d_in[0];
    const int*   parIdx   = (const int*)d_in[1];
    const float* omega    = (const float*)d_in[2];
    const float* childEmb = (const float*)d_in[3];
    const float* res      = (const float*)d_in[4];
    const float* leaves   = (const float*)d_in[5];
    const float* layerD   = (const float*)d_in[6];
    float* out = (float*)d_out;

    float* w        = (float*)d_ws;
    float* acc      = w;                 // 8 floats
    float* colSum   = w + 8;             // 1024
    float* inner    = w + 8 + NN;        // 1024
    float* omegaMat = w + 4096;          // 1024*1024

    init_kernel<<<5, 256, 0, stream>>>(w);
    node_loss_kernel<<<(NN * DD) / 256, 256, 0, stream>>>(idIdx, parIdx, childEmb, res, leaves, acc);
    inner_kernel<<<NN / 256, 256, 0, stream>>>(omega, inner);
    omega_dist_kernel<<<dim3(NN / 16, NN / 16), 32, 0, stream>>>(omega, inner, omegaMat, acc);
    pair_kernel<<<dim3(NN / 16, NN / 16), 256, 0, stream>>>(idIdx, childEmb, layerD, colSum, acc);
    dist_norm_kernel<<<256, 256, 0, stream>>>(omegaMat, colSum, acc);
    final_kernel<<<1, 32, 0, stream>>>(acc, out);
}